// SerialResnet_49933289783799
// MI455X (gfx1250) — compile-verified
//
#include <hip/hip_runtime.h>

#define NLAYERS 20

typedef __attribute__((ext_vector_type(2))) float v2f;
typedef __attribute__((ext_vector_type(8))) float v8f;

// gfx1250 hardware tanh (single trans op). Builtin when available, else raw ISA
// with a trailing v_nop to satisfy the TRANS co-execution hazard rule.
__device__ __forceinline__ float fast_tanh(float v) {
#if defined(__has_builtin)
#if __has_builtin(__builtin_amdgcn_tanhf)
  return __builtin_amdgcn_tanhf(v);
#else
  float r;
  asm volatile("v_tanh_f32 %0, %1\n\tv_nop" : "=v"(r) : "v"(v));
  return r;
#endif
#else
  float r;
  asm volatile("v_tanh_f32 %0, %1\n\tv_nop" : "=v"(r) : "v"(v));
  return r;
#endif
}

// Serial residual net, WIDTH=2, via V_WMMA_F32_16X16X4_F32.
// A (16x4) holds two block-diagonal copies of the 2x2 layer weight:
//   A[0:2, 0:2] = W_l   -> serves batch elements in lanes 0..15 (D rows M=0,1)
//   A[8:10,2:4] = W_l   -> serves batch elements in lanes 16..31 (D rows M=8,9)
// With the CDNA5 f32 layouts, lane n of B vgpr0/vgpr1 = (h0,h1) of element n
// and lane n of D vgpr0/vgpr1 = (pre0,pre1) of element n: no cross-lane moves.
// C is the inline-0 SRC2 operand; the per-layer bias is folded in with a
// v_add_f32 against an SGPR-resident scalar before the tanh.
template <int NG>
__global__ __launch_bounds__(256) void serial_resnet_wmma(
    const float* __restrict__ x, const float* __restrict__ W,
    const float* __restrict__ b, float* __restrict__ out, int n) {
  const unsigned lane = threadIdx.x & 31u;

  // ---- one-time per-wave prologue: build per-layer A matrices
  v2f A[NLAYERS];  // A vgpr0 (K=0 | K=2), A vgpr1 (K=1 | K=3) per ISA layout
  const bool row0 = (lane == 0u) | (lane == 24u);  // A[0,0]/A[0,1], A[8,2]/A[8,3]
  const bool row1 = (lane == 1u) | (lane == 25u);  // A[1,0]/A[1,1], A[9,2]/A[9,3]
#pragma unroll
  for (int l = 0; l < NLAYERS; ++l) {
    const float w00 = W[4 * l + 0], w01 = W[4 * l + 1];
    const float w10 = W[4 * l + 2], w11 = W[4 * l + 3];
    float a0 = 0.0f, a1 = 0.0f;
    if (row0) { a0 = w00; a1 = w01; }
    if (row1) { a0 = w10; a1 = w11; }
    A[l].x = a0;
    A[l].y = a1;
  }

  const long long CHUNK = 32ll * NG;  // elements per wave iteration
  const unsigned nthreads = gridDim.x * blockDim.x;
  const unsigned nwaves = nthreads >> 5;
  const unsigned wave = (blockIdx.x * blockDim.x + threadIdx.x) >> 5;

  // ---- main loop: wave-uniform control flow (EXEC all-1s for WMMA)
  for (long long base = (long long)wave * CHUNK; base + CHUNK <= (long long)n;
       base += (long long)nwaves * CHUNK) {
    float h0[NG], h1[NG];
#pragma unroll
    for (int g = 0; g < NG; ++g) {
      const float xv = x[base + 32ll * g + lane];  // coalesced b32 load
      h0[g] = xv;  // repeat_interleave: h = [x, x]
      h1[g] = xv;
    }
#pragma unroll
    for (int l = 0; l < NLAYERS; ++l) {
      // Uniform loads -> SGPRs (hoisted by the compiler); one scalar operand
      // per v_add_f32 is legal on the constant bus.
      const float b0 = b[2 * l + 0];
      const float b1 = b[2 * l + 1];
#pragma unroll
      for (int g = 0; g < NG; ++g) {
        v2f bm;
        bm.x = h0[g];  // B rows K=0 (lanes 0-15) / K=2 (lanes 16-31)
        bm.y = h1[g];  // B rows K=1 / K=3
        const v8f zero = {};  // lowers to inline SRC2=0 (no VGPRs)
        v8f d = __builtin_amdgcn_wmma_f32_16x16x4_f32(
            false, A[l], false, bm, (short)0, zero, false, false);
        // residual update, DT == 1.0
        h0[g] += fast_tanh(d[0] + b0);
        h1[g] += fast_tanh(d[1] + b1);
      }
    }
#pragma unroll
    for (int g = 0; g < NG; ++g)
      out[base + 32ll * g + lane] = 0.5f * (h0[g] + h1[g]);
  }

  // ---- scalar tail for n not a multiple of CHUNK (plain VALU path)
  const long long rem_start = ((long long)n / CHUNK) * CHUNK;
  for (long long i = rem_start + (blockIdx.x * blockDim.x + threadIdx.x);
       i < (long long)n; i += (long long)nthreads) {
    float h0 = x[i], h1 = h0;
#pragma unroll
    for (int l = 0; l < NLAYERS; ++l) {
      const float p0 = fmaf(h0, W[4 * l + 0], fmaf(h1, W[4 * l + 1], b[2 * l + 0]));
      const float p1 = fmaf(h0, W[4 * l + 2], fmaf(h1, W[4 * l + 3], b[2 * l + 1]));
      h0 += fast_tanh(p0);
      h1 += fast_tanh(p1);
    }
    out[i] = 0.5f * (h0 + h1);
  }
}

extern "C" void kernel_launch(void* const* d_in, const int* in_sizes, int n_in,
                              void* d_out, int out_size, void* d_ws, size_t ws_size,
                              hipStream_t stream) {
  const float* x = (const float*)d_in[0];
  const float* W = (const float*)d_in[1];  // [NLAYERS, 2, 2] row-major
  const float* b = (const float*)d_in[2];  // [NLAYERS, 2]
  float* out = (float*)d_out;
  const int n = in_sizes[0];  // BATCH (x is [B,1])

  // 2048 blocks x 256 threads = 16384 waves; at B=2^23 each wave runs 4
  // iterations of 128 elements -> prologue (A-matrix build) well amortized.
  const int threads = 256;
  const int blocks = 2048;
  serial_resnet_wmma<4><<<blocks, threads, 0, stream>>>(x, W, b, out, n);
}